// BeliefPropagation_21663815041320
// MI455X (gfx1250) — compile-verified
//
#include <hip/hip_runtime.h>

typedef __attribute__((ext_vector_type(2))) float v2f;
typedef __attribute__((ext_vector_type(4))) float v4f;
typedef __attribute__((ext_vector_type(8))) float v8f;

#define Q 8
#define TB 256

// ---------------- utility ----------------
__global__ void fill_zero_kernel(float* __restrict__ p, int n) {
  int i = blockIdx.x * blockDim.x + threadIdx.x;
  if (i < n) p[i] = 0.0f;
}

// deg[dst[e]] += 1  (segment_sum of ones)
__global__ void deg_kernel(const int* __restrict__ dst, float* __restrict__ deg, int E) {
  int e = blockIdx.x * blockDim.x + threadIdx.x;
  if (e < E) atomicAdd(&deg[dst[e]], 1.0f);
}

// ---------------- per-edge: log_src + segment-sum into node_logprod ----------------
__global__ void __launch_bounds__(TB)
edge_logsrc_kernel(const float* __restrict__ msg,
                   const float* __restrict__ edge_attr,
                   const int* __restrict__ dst,
                   const float* __restrict__ beta_p,
                   float* __restrict__ log_src,
                   float* __restrict__ nlp,
                   int E) {
  int e = blockIdx.x * blockDim.x + threadIdx.x;
  if (e >= E) return;
  float beta = beta_p[0];
  float ew = __expf(beta * edge_attr[e]) - 1.0f;

  const v4f* mrow = (const v4f*)(msg + (size_t)e * Q);
  v4f m0 = mrow[0];
  v4f m1 = mrow[1];
  v4f l0, l1;
#pragma unroll
  for (int k = 0; k < 4; ++k) l0[k] = log1pf(m0[k] * ew);
#pragma unroll
  for (int k = 0; k < 4; ++k) l1[k] = log1pf(m1[k] * ew);

  v4f* lrow = (v4f*)(log_src + (size_t)e * Q);
  lrow[0] = l0;
  lrow[1] = l1;

  float* nrow = nlp + (size_t)dst[e] * Q;  // 1.6MB array -> L2-resident atomics
#pragma unroll
  for (int k = 0; k < 4; ++k) atomicAdd(&nrow[k], l0[k]);
#pragma unroll
  for (int k = 0; k < 4; ++k) atomicAdd(&nrow[4 + k], l1[k]);
}

// ---------------- per-edge: new_msg = normalize(exp(h + nlp[src] - log_src[rev])) ----------------
__global__ void __launch_bounds__(TB)
edge_update_kernel(const int* __restrict__ src,
                   const float* __restrict__ nlp,
                   const float* __restrict__ log_src,
                   const float* __restrict__ h,
                   float* __restrict__ msg,
                   int E, int M) {
  int e = blockIdx.x * blockDim.x + threadIdx.x;
  if (e >= E) return;
  int r = (e < M) ? (e + M) : (e - M);  // contiguous gather (offset by +-M)
  const v4f* np = (const v4f*)(nlp + (size_t)src[e] * Q);
  const v4f* lp = (const v4f*)(log_src + (size_t)r * Q);
  const v4f* hp = (const v4f*)h;

  v4f a0 = hp[0] + np[0] - lp[0];
  v4f a1 = hp[1] + np[1] - lp[1];
  v4f t0, t1;
  float s = 0.0f;
#pragma unroll
  for (int k = 0; k < 4; ++k) { t0[k] = __expf(a0[k]); s += t0[k]; }
#pragma unroll
  for (int k = 0; k < 4; ++k) { t1[k] = __expf(a1[k]); s += t1[k]; }
  float inv = 1.0f / s;

  v4f* mrow = (v4f*)(msg + (size_t)e * Q);
  mrow[0] = t0 * inv;
  mrow[1] = t1 * inv;
}

// ---------------- per-node: new_psi = normalize(exp(h + nlp)) ----------------
__global__ void __launch_bounds__(TB)
node_update_kernel(const float* __restrict__ nlp,
                   const float* __restrict__ h,
                   float* __restrict__ psi, int N) {
  int i = blockIdx.x * blockDim.x + threadIdx.x;
  if (i >= N) return;
  const v4f* np = (const v4f*)(nlp + (size_t)i * Q);
  const v4f* hp = (const v4f*)h;
  v4f a0 = hp[0] + np[0];
  v4f a1 = hp[1] + np[1];
  v4f t0, t1;
  float s = 0.0f;
#pragma unroll
  for (int k = 0; k < 4; ++k) { t0[k] = __expf(a0[k]); s += t0[k]; }
#pragma unroll
  for (int k = 0; k < 4; ++k) { t1[k] = __expf(a1[k]); s += t1[k]; }
  float inv = 1.0f / s;
  v4f* prow = (v4f*)(psi + (size_t)i * Q);
  prow[0] = t0 * inv;
  prow[1] = t1 * inv;
}

// ---------------- field reduction via WMMA: h_acc[k] += sum_i deg[i]*psi[i][k] ----------------
// V_WMMA_F32_16X16X4_F32 (M=16,N=16,K=4), 4 nodes folded per issue.
// A (16x4 f32): lane half hf, element j -> K = j + 2*hf, value deg[node] (replicated over M).
// B (4x16 f32): column n = lane%16, element j -> K = j + 2*hf, value psi[node][n]
//   for n < 8; columns 8..15 zeroed via a multiplicative mask (no EXEC branches).
// C row M=0 = VGPR0 of lanes 0..15 -> lanes 0..7 carry the 8 group sums.
__global__ void __launch_bounds__(TB)
field_wmma_kernel(const float* __restrict__ deg,
                  const float* __restrict__ psi,
                  float* __restrict__ h_acc, int N) {
  int lane = threadIdx.x & 31;
  int hf = lane >> 4;
  int n = lane & 15;
  int wave = __builtin_amdgcn_readfirstlane(
      (int)((blockIdx.x * blockDim.x + threadIdx.x) >> 5));
  int nwaves = __builtin_amdgcn_readfirstlane((int)((gridDim.x * blockDim.x) >> 5));

  const float colmask = (n < Q) ? 1.0f : 0.0f;  // zero columns 8..15
  const int col = n & (Q - 1);                  // clamped column (always in range)
  const int k0 = 2 * hf;

  v8f c = {};
  int nfull = N >> 2;  // chunks with all 4 nodes in range -> branch-free body
  for (int ch = wave; ch < nfull; ch += nwaves) {
    int n0 = ch * 4 + k0;
    float a0 = deg[n0];
    float a1 = deg[n0 + 1];
    float b0 = colmask * psi[(size_t)n0 * Q + col];
    float b1 = colmask * psi[(size_t)(n0 + 1) * Q + col];
    v2f A = {a0, a1};
    v2f B = {b0, b1};
    // (neg_a, A, neg_b, B, c_mod, C, reuse_a, reuse_b)
    c = __builtin_amdgcn_wmma_f32_16x16x4_f32(false, A, false, B, (short)0, c,
                                              false, false);
  }

  // tail chunk (N % 4 != 0): wave 0 only; clamped indices + multiplicative masks,
  // EXEC stays all-ones at the WMMA.
  if (wave == 0 && (N & 3)) {
    int n0 = (nfull << 2) + k0;
    int n1 = n0 + 1;
    int i0 = (n0 < N) ? n0 : (N - 1);
    int i1 = (n1 < N) ? n1 : (N - 1);
    float m0 = (n0 < N) ? 1.0f : 0.0f;
    float m1 = (n1 < N) ? 1.0f : 0.0f;
    float a0 = m0 * deg[i0];
    float a1 = m1 * deg[i1];
    float b0 = m0 * colmask * psi[(size_t)i0 * Q + col];
    float b1 = m1 * colmask * psi[(size_t)i1 * Q + col];
    v2f A = {a0, a1};
    v2f B = {b0, b1};
    c = __builtin_amdgcn_wmma_f32_16x16x4_f32(false, A, false, B, (short)0, c,
                                              false, false);
  }

  if (lane < Q) atomicAdd(&h_acc[lane], c[0]);
}

__global__ void finalize_h_kernel(const float* __restrict__ h_acc,
                                  const float* __restrict__ beta_p,
                                  float* __restrict__ h, float invN) {
  int k = threadIdx.x;
  if (k < Q) h[k] = -beta_p[0] * invN * h_acc[k];
}

// ---------------- driver ----------------
extern "C" void kernel_launch(void* const* d_in, const int* in_sizes, int n_in,
                              void* d_out, int out_size, void* d_ws, size_t ws_size,
                              hipStream_t stream) {
  const int* edge_index = (const int*)d_in[0];   // [2, E]
  const float* edge_attr = (const float*)d_in[1];
  const float* message_init = (const float*)d_in[2];  // [E, 8]
  const float* psi_init = (const float*)d_in[3];      // [N, 8]
  const float* beta_p = (const float*)d_in[4];
  // d_in[5] = num_nodes (derived from sizes instead; capture-safe)

  const int E = in_sizes[2] / Q;
  const int N = in_sizes[3] / Q;
  const int M = E / 2;
  const int* src = edge_index;
  const int* dst = edge_index + E;

  // live state lives directly in d_out: [msg (E*Q) | psi (N*Q)]
  float* msg = (float*)d_out;
  float* psi = (float*)d_out + (size_t)E * Q;

  // workspace layout (all 16B aligned: E*Q, N*Q, N are multiples of 4)
  float* log_src = (float*)d_ws;                  // E*Q
  float* nlp = log_src + (size_t)E * Q;           // N*Q
  float* deg = nlp + (size_t)N * Q;               // N
  float* h_acc = deg + ((N + 3) & ~3);            // Q (keep 16B alignment)
  float* h = h_acc + Q;                           // Q

  hipMemcpyAsync(msg, message_init, sizeof(float) * (size_t)E * Q,
                 hipMemcpyDeviceToDevice, stream);
  hipMemcpyAsync(psi, psi_init, sizeof(float) * (size_t)N * Q,
                 hipMemcpyDeviceToDevice, stream);

  const int gE = (E + TB - 1) / TB;
  const int gN = (N + TB - 1) / TB;
  const int gNQ = (N * Q + TB - 1) / TB;
  const float invN = 1.0f / (float)N;

  // degree (fixed across iterations)
  fill_zero_kernel<<<gN, TB, 0, stream>>>(deg, N);
  deg_kernel<<<gE, TB, 0, stream>>>(dst, deg, E);

  // h0 = field_of(psi_init)
  fill_zero_kernel<<<1, 32, 0, stream>>>(h_acc, Q);
  field_wmma_kernel<<<64, TB, 0, stream>>>(deg, psi, h_acc, N);
  finalize_h_kernel<<<1, 32, 0, stream>>>(h_acc, beta_p, h, invN);

  for (int it = 0; it < 10; ++it) {
    fill_zero_kernel<<<gNQ, TB, 0, stream>>>(nlp, N * Q);
    edge_logsrc_kernel<<<gE, TB, 0, stream>>>(msg, edge_attr, dst, beta_p,
                                              log_src, nlp, E);
    edge_update_kernel<<<gE, TB, 0, stream>>>(src, nlp, log_src, h, msg, E, M);
    node_update_kernel<<<gN, TB, 0, stream>>>(nlp, h, psi, N);
    // h = field_of(new_psi) for next iteration
    fill_zero_kernel<<<1, 32, 0, stream>>>(h_acc, Q);
    field_wmma_kernel<<<64, TB, 0, stream>>>(deg, psi, h_acc, N);
    finalize_h_kernel<<<1, 32, 0, stream>>>(h_acc, beta_p, h, invN);
  }
}